// YoloLayer_45414984188538
// MI455X (gfx1250) — compile-verified
//
#include <hip/hip_runtime.h>
#include <hip/hip_bf16.h>
#include <stdint.h>

// ---------------- problem constants ----------------
#define NUM_B   16
#define NUM_A   3
#define HH      76
#define WW      76
#define HWTOT   (HH*WW)              // 5776
#define NCLS    80
#define NCH     85                   // 5 + 80 channels per anchor
#define TILE    152                  // locations per block (2 image rows), 5776 = 38*152
#define PITCH   156                  // LDS pitch in dwords (mult of 4 -> 16B rows; 2-way-max bank conflicts)
#define TILES_PER_BA (HWTOT/TILE)    // 38
#define NTHREADS 160                 // 5 waves (wave32)
#define NLOC_PER_B (NUM_A*HWTOT)     // 17328
#define BOX_FLOATS (NUM_B*NLOC_PER_B*4)  // 1108992 (boxes precede confs in d_out)

#define L2E 1.44269504088896340736f

__device__ __forceinline__ float fexp2(float x) { return __builtin_amdgcn_exp2f(x); }   // v_exp_f32
__device__ __forceinline__ float frcp(float x)  { return __builtin_amdgcn_rcpf(x); }    // v_rcp_f32
__device__ __forceinline__ float fexp(float x)  { return fexp2(L2E * x); }
__device__ __forceinline__ float fsigmoid(float x) { return frcp(1.0f + fexp2(-L2E * x)); }

// CDNA5 async tensor-path: per-lane 16B global -> LDS DMA, tracked by ASYNCcnt.
__device__ __forceinline__ void async_load_b128(uint32_t lds_byte_addr, const void* gptr) {
    asm volatile("global_load_async_to_lds_b128 %0, %1, off"
                 :: "v"(lds_byte_addr), "v"((uint64_t)(uintptr_t)gptr)
                 : "memory");
}
__device__ __forceinline__ void wait_async0() {
    asm volatile("s_wait_asynccnt 0" ::: "memory");
}

__global__ __launch_bounds__(NTHREADS)
void yolo_head_kernel(const float* __restrict__ src,
                      float* __restrict__ boxes,
                      float* __restrict__ confs)
{
    __shared__ float smem[NCH * PITCH];   // 85*156*4 = 53,040 B
    __shared__ float scale_s[TILE];

    const int t    = threadIdx.x;
    const int tile = blockIdx.x % TILES_PER_BA;
    const int ba   = blockIdx.x / TILES_PER_BA;   // b*3 + a
    const int a    = ba % NUM_A;
    const int b    = ba / NUM_A;
    const int hw0  = tile * TILE;

    const float* base = src + (size_t)ba * NCH * HWTOT + hw0;

    // ---- Phase 1: async DMA the (85 x 152) fp32 slab into LDS ----
    // 85 rows * 38 float4-chunks = 3230 b128 transfers. All 16B aligned:
    // global dword off = c*5776 + hw0 + 4k (all multiples of 4); LDS row pitch 156 dwords.
    const uint32_t lds_base = (uint32_t)(uintptr_t)(&smem[0]);  // flat addr low 32 bits == LDS byte offset
    #pragma unroll 1
    for (int q = t; q < NCH * (TILE / 4); q += NTHREADS) {
        int c  = q / (TILE / 4);
        int j4 = (q - c * (TILE / 4)) * 4;
        async_load_b128(lds_base + (uint32_t)(c * PITCH + j4) * 4u,
                        base + (size_t)c * HWTOT + j4);
    }
    wait_async0();
    __syncthreads();

    // ---- Phase 2: one thread per location ----
    if (t < TILE) {
        const int ln = t;
        const int hw = hw0 + ln;
        const int h  = hw / WW;
        const int w  = hw - h * WW;
        const float invW = 1.0f / (float)WW;
        const float invH = 1.0f / (float)HH;
        // ANCHOR_{W,H}/STRIDE divided by W(=H)=76, selected by anchor a
        const float AW[NUM_A] = {1.5f / 76.0f, 2.375f / 76.0f, 5.0f / 76.0f};
        const float AH[NUM_A] = {2.0f / 76.0f, 4.5f / 76.0f, 3.5f / 76.0f};

        float x0 = smem[0 * PITCH + ln];
        float x1 = smem[1 * PITCH + ln];
        float x2 = smem[2 * PITCH + ln];
        float x3 = smem[3 * PITCH + ln];
        float x4 = smem[4 * PITCH + ln];

        float bx  = (fsigmoid(x0) + (float)w) * invW;
        float by  = (fsigmoid(x1) + (float)h) * invH;
        float bw  = fexp(x2) * AW[a];
        float bh  = fexp(x3) * AH[a];
        float det = fsigmoid(x4);

        const int n = a * HWTOT + hw;
        *(float4*)(boxes + ((size_t)b * NLOC_PER_B + n) * 4) = make_float4(bx, by, bw, bh);

        // stable softmax over 80 classes; unnormalized exp written back in place (scaled by det/sum at writeout)
        float mx = -3.0e38f;
        #pragma unroll 8
        for (int c = 0; c < NCLS; ++c)
            mx = fmaxf(mx, smem[(5 + c) * PITCH + ln]);
        float sum = 0.0f;
        #pragma unroll 8
        for (int c = 0; c < NCLS; ++c) {
            float e = fexp2(L2E * (smem[(5 + c) * PITCH + ln] - mx));
            smem[(5 + c) * PITCH + ln] = e;
            sum += e;
        }
        scale_s[ln] = det * frcp(sum);
    }
    __syncthreads();

    // ---- Phase 3: transposed, fully-coalesced confs writeout ----
    // 152 loc * 20 float4 = 3040 chunks = exactly 19 per thread.
    const size_t conf_base = ((size_t)b * NLOC_PER_B + (size_t)a * HWTOT + hw0) * NCLS;
    #pragma unroll 1
    for (int e = t; e < TILE * (NCLS / 4); e += NTHREADS) {
        int ln = e / (NCLS / 4);
        int c4 = (e - ln * (NCLS / 4)) * 4;
        float s = scale_s[ln];
        float4 o;
        o.x = smem[(5 + c4 + 0) * PITCH + ln] * s;
        o.y = smem[(5 + c4 + 1) * PITCH + ln] * s;
        o.z = smem[(5 + c4 + 2) * PITCH + ln] * s;
        o.w = smem[(5 + c4 + 3) * PITCH + ln] * s;
        *(float4*)(confs + conf_base + (size_t)ln * NCLS + c4) = o;
    }
}

extern "C" void kernel_launch(void* const* d_in, const int* in_sizes, int n_in,
                              void* d_out, int out_size, void* d_ws, size_t ws_size,
                              hipStream_t stream)
{
    const float* src = (const float*)d_in[0];
    float* out   = (float*)d_out;
    float* boxes = out;                 // (16, 17328, 4)
    float* confs = out + BOX_FLOATS;    // (16, 17328, 80)

    dim3 grid(NUM_B * NUM_A * TILES_PER_BA);   // 1824 blocks
    dim3 block(NTHREADS);
    yolo_head_kernel<<<grid, block, 0, stream>>>(src, boxes, confs);
}